// HierarchicalEncoder_74586402063031
// MI455X (gfx1250) — compile-verified
//
#include <hip/hip_runtime.h>
#include <math.h>

// ---------------------------------------------------------------------------
// CDNA5 WMMA f32 16x16x4:  A = 16x4 f32 (2 VGPR), B = 4x16 f32 (2 VGPR),
// C/D = 16x16 f32 (8 VGPR).  Fragment mapping (ISA 7.12.2):
//   A: lane half h, component c -> K = 2h+c, M = lane%16
//   B: symmetric with N = lane%16
//   C: row = vgpr + 8*(lane/16), col = lane%16
// ---------------------------------------------------------------------------
typedef float v2f __attribute__((ext_vector_type(2)));
typedef float v8f __attribute__((ext_vector_type(8)));

__device__ __forceinline__ v8f wmma4(v2f a, v2f b, v8f c) {
  return __builtin_amdgcn_wmma_f32_16x16x4_f32(false, a, false, b, (short)0, c,
                                               false, false);
}

__device__ __forceinline__ v8f vzero8() {
  v8f r;
#pragma unroll
  for (int i = 0; i < 8; ++i) r[i] = 0.f;
  return r;
}

__device__ __forceinline__ float gelu_f(float x) {
  return 0.5f * x * (1.f + erff(x * 0.70710678118654752f));
}

// Async global -> LDS copy (16B per active lane), tracked by ASYNCcnt.
__device__ __forceinline__ void async_load_b128(unsigned lds_off,
                                                const float* gaddr) {
  asm volatile("global_load_async_to_lds_b128 %0, %1, off"
               :
               : "v"(lds_off), "v"(gaddr)
               : "memory");
}

__device__ __forceinline__ void wait_asynccnt0() {
  asm volatile("s_wait_asynccnt 0x0" ::: "memory");
}

// ---------------------------------------------------------------------------
// Tiled GEMM: C[M,N] = act( scale(row) * (A[M,K] @ W[K,N]) + bias[N] )
// Block tile 128x64 (256 threads = 8 waves, 4x2 wave grid, 32x32 per wave).
// K staged in 16-wide double-buffered LDS tiles filled by async b128 loads.
// N multiple of 64, K multiple of 16 (true for every shape here).
// ---------------------------------------------------------------------------
template <int ACT, bool SCALE>
__global__ __launch_bounds__(256) void gemm_kernel(
    const float* __restrict__ A, const float* __restrict__ W,
    const float* __restrict__ bias, const float* __restrict__ rowscale,
    int scale_stride, float* __restrict__ C, int M, int N, int K) {
  // As k-minor, Bs n-minor; row strides are multiples of 16B for async b128 stores.
  __shared__ float As[2][128][20];  // [buf][m][k] (16 k + 4 pad)
  __shared__ float Bs[2][16][68];   // [buf][k][n] (64 n + 4 pad)

  const int t = threadIdx.x;
  const int lane = t & 31;
  const int w = t >> 5;   // 0..7
  const int wm = w >> 1;  // 0..3  (M direction)
  const int wn = w & 1;   // 0..1  (N direction)
  const int m0 = blockIdx.y * 128;
  const int n0 = blockIdx.x * 64;
  const int lm = lane & 15;
  const int lh = lane >> 4;

  v8f acc[2][2];
  acc[0][0] = vzero8();
  acc[0][1] = vzero8();
  acc[1][0] = vzero8();
  acc[1][1] = vzero8();

  // Issue one 16-wide K tile (A: 128x16, B: 16x64) into buffer pb.
  auto issue = [&](int k0, int pb) {
#pragma unroll
    for (int j = 0; j < 2; ++j) {
      const int chunk = t + j * 256;      // 0..511
      const int row = chunk >> 2;         // 0..127
      const int kc = (chunk & 3) * 4;     // 0,4,8,12
      int gm = m0 + row;
      if (gm > M - 1) gm = M - 1;         // clamp: always-valid reads
      const float* ga = A + (long)gm * K + k0 + kc;
      async_load_b128((unsigned)(size_t)&As[pb][row][kc], ga);
    }
    if (t < 64) {
      const int krow = t >> 2;            // 0..15
      const int nc = (t & 3) * 4;         // 0,4,8,12 -> *? (16 floats/row chunk)
      const float* gb = W + (long)(k0 + krow) * N + n0 + nc * 4;
      async_load_b128((unsigned)(size_t)&Bs[pb][krow][nc * 4], gb);
    }
  };

  issue(0, 0);
  int buf = 0;
  for (int k0 = 0; k0 < K; k0 += 16) {
    wait_asynccnt0();
    __syncthreads();
    if (k0 + 16 < K) issue(k0 + 16, buf ^ 1);
#pragma unroll
    for (int kk = 0; kk < 16; kk += 4) {
      v2f a0, a1, b0, b1;
      a0.x = As[buf][wm * 32 + lm][kk + 2 * lh + 0];
      a0.y = As[buf][wm * 32 + lm][kk + 2 * lh + 1];
      a1.x = As[buf][wm * 32 + 16 + lm][kk + 2 * lh + 0];
      a1.y = As[buf][wm * 32 + 16 + lm][kk + 2 * lh + 1];
      b0.x = Bs[buf][kk + 2 * lh + 0][wn * 32 + lm];
      b0.y = Bs[buf][kk + 2 * lh + 1][wn * 32 + lm];
      b1.x = Bs[buf][kk + 2 * lh + 0][wn * 32 + 16 + lm];
      b1.y = Bs[buf][kk + 2 * lh + 1][wn * 32 + 16 + lm];
      acc[0][0] = wmma4(a0, b0, acc[0][0]);
      acc[0][1] = wmma4(a0, b1, acc[0][1]);
      acc[1][0] = wmma4(a1, b0, acc[1][0]);
      acc[1][1] = wmma4(a1, b1, acc[1][1]);
    }
    buf ^= 1;
  }

#pragma unroll
  for (int im = 0; im < 2; ++im) {
#pragma unroll
    for (int v = 0; v < 8; ++v) {
      const int row = m0 + wm * 32 + im * 16 + v + 8 * lh;
      if (row >= M) continue;
      float sc = 1.f;
      if (SCALE) sc = rowscale[(long)row * scale_stride];
      const int col0 = n0 + wn * 32 + lm;
      const int col1 = col0 + 16;
      float r0 = acc[im][0][v] * sc + (bias ? bias[col0] : 0.f);
      float r1 = acc[im][1][v] * sc + (bias ? bias[col1] : 0.f);
      if (ACT == 1) { r0 = fmaxf(r0, 0.f); r1 = fmaxf(r1, 0.f); }
      if (ACT == 2) { r0 = gelu_f(r0); r1 = gelu_f(r1); }
      C[(long)row * N + col0] = r0;
      C[(long)row * N + col1] = r1;
    }
  }
}

// ---------------------------------------------------------------------------
// N==3 GEMM (pos-pred head, composer gates). One thread per row.
// ---------------------------------------------------------------------------
template <bool SOFTMAX>
__global__ __launch_bounds__(256) void gemm3_kernel(
    const float* __restrict__ A, const float* __restrict__ W,
    const float* __restrict__ b3, float* __restrict__ out, int M, int K) {
  int m = blockIdx.x * 256 + threadIdx.x;
  if (m >= M) return;
  float a0 = b3[0], a1 = b3[1], a2 = b3[2];
  const float* ar = A + (long)m * K;
  for (int k = 0; k < K; ++k) {
    float av = ar[k];
    a0 += av * W[k * 3 + 0];
    a1 += av * W[k * 3 + 1];
    a2 += av * W[k * 3 + 2];
  }
  if (SOFTMAX) {
    float mx = fmaxf(a0, fmaxf(a1, a2));
    float e0 = __expf(a0 - mx), e1 = __expf(a1 - mx), e2 = __expf(a2 - mx);
    float s = e0 + e1 + e2;
    a0 = e0 / s; a1 = e1 / s; a2 = e2 / s;
  }
  out[(long)m * 3 + 0] = a0;
  out[(long)m * 3 + 1] = a1;
  out[(long)m * 3 + 2] = a2;
}

// ---------------------------------------------------------------------------
// Per-batch: softmax(jp), starts, inclusive cumsum -> idx, seg, max idx.
// ---------------------------------------------------------------------------
__global__ __launch_bounds__(1024) void seg_kernel(
    const float* __restrict__ jp, float* __restrict__ pps,
    int* __restrict__ seg, float* __restrict__ idx_out,
    int* __restrict__ maxidx, int L) {
  __shared__ float sc[1024];
  __shared__ int mxs[1024];
  const int b = blockIdx.x, l = threadIdx.x;
  const long o = (long)b * L + l;
  float j0 = jp[o * 3 + 0], j1 = jp[o * 3 + 1], j2 = jp[o * 3 + 2];
  float mx = fmaxf(j0, fmaxf(j1, j2));
  float e0 = __expf(j0 - mx), e1 = __expf(j1 - mx), e2 = __expf(j2 - mx);
  float s = e0 + e1 + e2;
  float p0 = e0 / s;
  pps[o * 3 + 0] = p0;
  pps[o * 3 + 1] = e1 / s;
  pps[o * 3 + 2] = e2 / s;
  float start = (p0 > 0.5f) ? 1.f : 0.f;
  float nz = ((j0 + j1 + j2) == 0.f) ? 0.f : 1.f;
  sc[l] = start;
  __syncthreads();
  for (int off = 1; off < 1024; off <<= 1) {
    float v = (l >= off) ? sc[l - off] : 0.f;
    __syncthreads();
    sc[l] += v;
    __syncthreads();
  }
  float idxf = sc[l] * nz;
  idx_out[o] = idxf;
  seg[o] = (int)idxf - 1;
  mxs[l] = (int)idxf;
  __syncthreads();
  for (int off = 512; off > 0; off >>= 1) {
    if (l < off) mxs[l] = max(mxs[l], mxs[l + off]);
    __syncthreads();
  }
  if (l == 0) maxidx[b] = mxs[0];
}

// ---------------------------------------------------------------------------
// Segment counts and segment sums (f32 atomics).
// ---------------------------------------------------------------------------
__global__ __launch_bounds__(256) void count_kernel(const int* __restrict__ seg,
                                                    float* __restrict__ cnt,
                                                    int L, int S, int n) {
  int i = blockIdx.x * 256 + threadIdx.x;
  if (i >= n) return;
  int b = i / L;
  int s = seg[i];
  if (s >= 0 && s < S) atomicAdd(&cnt[(long)b * S + s], 1.f);
}

__global__ __launch_bounds__(256) void segsum_kernel(
    const float* __restrict__ feats, const int* __restrict__ seg,
    float* __restrict__ dst, int L, int S) {
  const int b = blockIdx.y, l = blockIdx.x;
  const int s = seg[(long)b * L + l];
  if (s < 0 || s >= S) return;
  const long src = ((long)b * L + l) * 512;
  const long d0 = ((long)b * S + s) * 512;
  for (int d = threadIdx.x; d < 512; d += 256)
    atomicAdd(&dst[d0 + d], feats[src + d]);
}

// ---------------------------------------------------------------------------
// Elementwise helpers.
// ---------------------------------------------------------------------------
__global__ __launch_bounds__(256) void comb_kernel(
    const float* __restrict__ si, const float* __restrict__ sm,
    const float* __restrict__ sf, float* __restrict__ comb, long total) {
  long i = (long)blockIdx.x * 256 + threadIdx.x;
  if (i >= total) return;
  long r = i / 1536;
  int k = (int)(i - r * 1536);
  float v = (k < 512) ? si[r * 512 + k]
            : (k < 1024) ? sm[r * 512 + (k - 512)]
                         : sf[r * 512 + (k - 1024)];
  comb[i] = v;
}

__global__ __launch_bounds__(256) void gate_kernel(float* __restrict__ comb,
                                                   const float* __restrict__ g,
                                                   long total) {
  long i = (long)blockIdx.x * 256 + threadIdx.x;
  if (i >= total) return;
  long r = i / 1536;
  int k = (int)(i - r * 1536);
  comb[i] *= g[r * 3 + (k >> 9)];
}

__global__ __launch_bounds__(256) void mask_kernel(float* __restrict__ syl,
                                                   const int* __restrict__ mxi,
                                                   int S, long total) {
  long i = (long)blockIdx.x * 256 + threadIdx.x;
  if (i >= total) return;
  long r = i / 512;
  int b = (int)(r / S);
  int s = (int)(r - (long)b * S);
  if (s > mxi[b]) syl[i] = 0.f;
}

__global__ __launch_bounds__(256) void gather_kernel(
    const float* __restrict__ jamo, const float* __restrict__ sylctx,
    const int* __restrict__ seg, float* __restrict__ out, int L, int S,
    long total) {
  long i = (long)blockIdx.x * 256 + threadIdx.x;
  if (i >= total) return;
  long r = i / 512;
  int d = (int)(i - r * 512);
  int b = (int)(r / L);
  int sg = seg[r];
  float add = 0.f;
  if (sg >= 0) {
    int sc = sg < (S - 1) ? sg : (S - 1);
    add = sylctx[((long)b * S + sc) * 512 + d];
  }
  out[i] = jamo[i] + add;
}

__global__ __launch_bounds__(256) void enhs_kernel(
    const float* __restrict__ syl, const float* __restrict__ sums,
    const float* __restrict__ cnt, float* __restrict__ out, long total) {
  long i = (long)blockIdx.x * 256 + threadIdx.x;
  if (i >= total) return;
  long r = i / 512;
  float c = cnt[r];
  float add = (c > 0.f) ? sums[i] / fmaxf(c, 1.f) : 0.f;
  out[i] = syl[i] + add;
}

// ---------------------------------------------------------------------------
// LayerNorm over D=512: out = ln(a (+ res)) * g + be.  Grid = M, block 256.
// ---------------------------------------------------------------------------
__global__ __launch_bounds__(256) void ln_kernel(
    const float* __restrict__ a, const float* __restrict__ res,
    const float* __restrict__ g, const float* __restrict__ be,
    float* __restrict__ out) {
  __shared__ float red[256];
  const int t = threadIdx.x;
  const long base = (long)blockIdx.x * 512;
  float v0 = a[base + t] + (res ? res[base + t] : 0.f);
  float v1 = a[base + t + 256] + (res ? res[base + t + 256] : 0.f);
  red[t] = v0 + v1;
  __syncthreads();
  for (int off = 128; off > 0; off >>= 1) {
    if (t < off) red[t] += red[t + off];
    __syncthreads();
  }
  float mean = red[0] * (1.f / 512.f);
  __syncthreads();
  float d0 = v0 - mean, d1 = v1 - mean;
  red[t] = d0 * d0 + d1 * d1;
  __syncthreads();
  for (int off = 128; off > 0; off >>= 1) {
    if (t < off) red[t] += red[t + off];
    __syncthreads();
  }
  float rs = rsqrtf(red[0] * (1.f / 512.f) + 1e-5f);
  out[base + t] = d0 * rs * g[t] + be[t];
  out[base + t + 256] = d1 * rs * g[t + 256] + be[t + 256];
}

// ---------------------------------------------------------------------------
// Fused flash attention, H=8 heads, DH=64, f32 WMMA for QK^T and P.V.
// Grid = (ceil(Lx/16), H, B); block = 32 (one wave).
// qkv layout: [b, l, 3*512] (q | k | v);  out layout: [b, l, 512].
// ---------------------------------------------------------------------------
__global__ __launch_bounds__(32) void attn_kernel(const float* __restrict__ qkv,
                                                  float* __restrict__ out,
                                                  int Lx) {
  __shared__ float Qs[64][17];  // [dh][q]
  __shared__ float Ks[64][17];  // [dh][key]
  __shared__ float Vs[16][65];  // [key][dh]
  __shared__ float Ps[16][17];  // [q][key]
  const int b = blockIdx.z, h = blockIdx.y;
  const int q0 = blockIdx.x * 16;
  const int t = threadIdx.x;
  const int lm = t & 15, lh = t >> 4;
  const long rstride = 1536;
  const float* base = qkv + (long)b * Lx * rstride + h * 64;

  for (int e = t; e < 1024; e += 32) {
    int r = e >> 6, k = e & 63;
    Qs[k][r] = (q0 + r < Lx) ? base[(long)(q0 + r) * rstride + k] : 0.f;
  }

  v8f o0 = vzero8(), o1 = vzero8(), o2 = vzero8(), o3 = vzero8();
  float rmax[8], rsum[8];
#pragma unroll
  for (int v = 0; v < 8; ++v) { rmax[v] = -3.0e38f; rsum[v] = 0.f; }

  for (int j0 = 0; j0 < Lx; j0 += 16) {
    __syncthreads();
    for (int e = t; e < 1024; e += 32) {
      int n = e & 15, k = e >> 4;
      bool ok = (j0 + n) < Lx;
      Ks[k][n] = ok ? base[(long)(j0 + n) * rstride + 512 + k] : 0.f;
    }
    for (int e = t; e < 1024; e += 32) {
      int n = e >> 6, k = e & 63;
      bool ok = (j0 + n) < Lx;
      Vs[n][k] = ok ? base[(long)(j0 + n) * rstride + 1024 + k] : 0.f;
    }
    __syncthreads();

    v8f sf = vzero8();
#pragma unroll
    for (int kk = 0; kk < 64; kk += 4) {
      v2f a, bb;
      a.x = Qs[kk + 2 * lh + 0][lm];
      a.y = Qs[kk + 2 * lh + 1][lm];
      bb.x = Ks[kk + 2 * lh + 0][lm];
      bb.y = Ks[kk + 2 * lh + 1][lm];
      sf = wmma4(a, bb, sf);
    }

    const bool colok = (j0 + lm) < Lx;
    float pv[8];
#pragma unroll
    for (int v = 0; v < 8; ++v) {
      float sv = colok ? sf[v] * 0.125f : -3.0e38f;
      float tm = sv;
      tm = fmaxf(tm, __shfl_xor(tm, 1, 32));
      tm = fmaxf(tm, __shfl_xor(tm, 2, 32));
      tm = fmaxf(tm, __shfl_xor(tm, 4, 32));
      tm = fmaxf(tm, __shfl_xor(tm, 8, 32));
      float nm = fmaxf(rmax[v], tm);
      float corr = __expf(rmax[v] - nm);
      float p = __expf(sv - nm);
      float ts = p;
      ts += __shfl_xor(ts, 1, 32);
      ts += __shfl_xor(ts, 2, 32);
      ts += __shfl_xor(ts, 4, 32);
      ts += __shfl_xor(ts, 8, 32);
      rsum[v] = rsum[v] * corr + ts;
      rmax[v] = nm;
      o0[v] *= corr; o1[v] *= corr; o2[v] *= corr; o3[v] *= corr;
      pv[v] = p;
    }
#pragma unroll
    for (int v = 0; v < 8; ++v) Ps[v + 8 * lh][lm] = pv[v];
    __syncthreads();
#pragma unroll
    for (int kk = 0; kk < 16; kk += 4) {
      v2f a;
      a.x = Ps[lm][kk + 2 * lh + 0];
      a.y = Ps[lm][kk + 2 * lh + 1];
      v2f b0, b1, b2, b3;
      b0.x = Vs[kk + 2 * lh + 0][lm];      b0.y = Vs[kk + 2 * lh + 1][lm];
      b1.x = Vs[kk + 2 * lh + 0][16 + lm]; b1.y = Vs[kk + 2 * lh + 1][16 + lm];
      b2.x = Vs[kk + 2 * lh + 0][32 + lm]; b2.y = Vs[kk + 2 * lh + 1][32 + lm];
      b3.x = Vs[kk + 2 * lh + 0][48 + lm]; b3.y = Vs[kk + 2 * lh + 1][48 + lm];
      o0 = wmma4(a, b0, o0);
      o1 = wmma4(a, b1, o1);
      o2 = wmma4(a, b2, o2);
      o3 = wmma4(a, b3, o3);
    }
  }

#pragma unroll
  for (int v = 0; v < 8; ++v) {
    int r = v + 8 * lh;
    if (q0 + r < Lx) {
      long ob = ((long)b * Lx + q0 + r) * 512 + h * 64;
      float inv = 1.f / rsum[v];
      out[ob + lm] = o0[v] * inv;
      out[ob + 16 + lm] = o1[v] * inv;
      out[ob + 32 + lm] = o2[v] * inv;
      out[ob + 48 + lm] = o3[v] * inv;
    }
  }
}

// ---------------------------------------------------------------------------
// Host-side orchestration.
// ---------------------------------------------------------------------------
static inline void launch_gemm(int act, const float* A, const float* W,
                               const float* bias, const float* rs, int rstride,
                               float* C, long M, int N, int K,
                               hipStream_t st) {
  dim3 g((unsigned)(N / 64), (unsigned)((M + 127) / 128));
  if (rs) {
    if (act == 0) gemm_kernel<0, true><<<g, 256, 0, st>>>(A, W, bias, rs, rstride, C, (int)M, N, K);
    else if (act == 1) gemm_kernel<1, true><<<g, 256, 0, st>>>(A, W, bias, rs, rstride, C, (int)M, N, K);
    else gemm_kernel<2, true><<<g, 256, 0, st>>>(A, W, bias, rs, rstride, C, (int)M, N, K);
  } else {
    if (act == 0) gemm_kernel<0, false><<<g, 256, 0, st>>>(A, W, bias, nullptr, 0, C, (int)M, N, K);
    else if (act == 1) gemm_kernel<1, false><<<g, 256, 0, st>>>(A, W, bias, nullptr, 0, C, (int)M, N, K);
    else gemm_kernel<2, false><<<g, 256, 0, st>>>(A, W, bias, nullptr, 0, C, (int)M, N, K);
  }
}

extern "C" void kernel_launch(void* const* d_in, const int* in_sizes, int n_in,
                              void* d_out, int out_size, void* d_ws,
                              size_t ws_size, hipStream_t stream) {
  (void)in_sizes; (void)n_in; (void)ws_size;
  const int B = 8, L = 1024, D = 512, F = 2048, NL = 2;
  const long BL = (long)B * L;

  // out = jamo[B,L,D] + syl[B,S,D] + jp[B,L,3] + idx[B,L]  -> recover S
  long S = ((long)out_size - BL * D - BL * 3 - BL) / ((long)B * D);
  if (S < 1) S = 1;
  const long BS = (long)B * S;
  const long rowsMax = (BL > BS) ? BL : BS;

  auto F32 = [&](int i) { return (const float*)d_in[i]; };
  const float* x = F32(0);
  const float *posW1 = F32(1), *posb1 = F32(2), *posW2 = F32(3), *posb2 = F32(4);
  const float *Wi = F32(5), *bi = F32(6), *Wm = F32(7), *bm = F32(8);
  const float *Wf = F32(9), *bf = F32(10), *Wc = F32(11), *bc = F32(12);
  const float *Wg = F32(13), *bg = F32(14);
  const int JL[2] = {15, 27}, SL[2] = {39, 51}, CR[2] = {63, 75};

  // workspace carve
  char* wp = (char*)d_ws;
  auto alloc = [&](size_t bytes) -> void* {
    void* r = (void*)wp;
    wp += (bytes + 255) & ~(size_t)255;
    return r;
  };
  float* jp = (float*)alloc(BL * 3 * 4);
  float* pps = (float*)alloc(BL * 3 * 4);
  int* seg = (int*)alloc(BL * 4);
  int* mxi = (int*)alloc(64 * 4);
  float* cnt = (float*)alloc(BS * 4);
  float* si = (float*)alloc(BS * 512 * 4);
  float* sm = (float*)alloc(BS * 512 * 4);
  float* sf = (float*)alloc(BS * 512 * 4);
  float* comb = (float*)alloc(BS * 1536 * 4);
  float* gates = (float*)alloc(BS * 3 * 4);
  float* jamo = (float*)alloc(BL * 512 * 4);
  float* syl = (float*)alloc(BS * 512 * 4);
  float* t0 = (float*)alloc(BL * 2048 * 4);
  float* t1 = (float*)alloc(rowsMax * 512 * 4);
  float* t2 = (float*)alloc(rowsMax * 512 * 4);

  float* outJamo = (float*)d_out;
  float* outSyl = outJamo + BL * D;
  float* outJp = outSyl + BS * D;
  float* outIdx = outJp + BL * 3;

  auto gridE = [](long n) { return dim3((unsigned)((n + 255) / 256)); };

  // ---- pos_pred(x) -> jp ----
  launch_gemm(2, x, posW1, posb1, nullptr, 0, t0, BL, 256, 512, stream);
  gemm3_kernel<false><<<gridE(BL), 256, 0, stream>>>(t0, posW2, posb2, jp, (int)BL, 256);

  // ---- segmentation ----
  hipMemsetAsync(mxi, 0, 64 * 4, stream);
  seg_kernel<<<B, 1024, 0, stream>>>(jp, pps, seg, outIdx, mxi, L);

  // ---- composer ----
  hipMemsetAsync(cnt, 0, BS * 4, stream);
  hipMemsetAsync(si, 0, BS * 512 * 4, stream);
  hipMemsetAsync(sm, 0, BS * 512 * 4, stream);
  hipMemsetAsync(sf, 0, BS * 512 * 4, stream);
  count_kernel<<<gridE(BL), 256, 0, stream>>>(seg, cnt, L, (int)S, (int)BL);

  dim3 gseg((unsigned)L, (unsigned)B);
  launch_gemm(0, x, Wi, bi, pps + 0, 3, t1, BL, 512, 512, stream);
  segsum_kernel<<<gseg, 256, 0, stream>>>(t1, seg, si, L, (int)S);
  launch_gemm(0, x, Wm, bm, pps + 1, 3, t1, BL, 512, 512, stream);
  segsum_kernel<<<gseg, 256, 0, stream>>>(t1, seg, sm, L, (int)S);
  launch_gemm(0, x, Wf, bf, pps + 2, 3, t1, BL, 512, 512, stream);
  segsum_kernel<<<gseg, 256, 0, stream>>>(t1, seg, sf, L, (int)S);

  comb_kernel<<<gridE(BS * 1536), 256, 0, stream>>>(si, sm, sf, comb, BS * 1536);
  gemm3_kernel<true><<<gridE(BS), 256, 0, stream>>>(comb, Wg, bg, gates, (int)BS, 1536);
  gate_kernel<<<gridE(BS * 1536), 256, 0, stream>>>(comb, gates, BS * 1536);
  launch_gemm(0, comb, Wc, bc, nullptr, 0, syl, BS, 512, 1536, stream);
  mask_kernel<<<gridE(BS * 512), 256, 0, stream>>>(syl, mxi, (int)S, BS * 512);

  hipMemcpyAsync(jamo, x, BL * 512 * 4, hipMemcpyDeviceToDevice, stream);

  // ---- encoder + cross layers ----
  auto enc = [&](float* xb, long M, int Lx, int pi) {
    const float *Wqkv = F32(pi + 0), *bqkv = F32(pi + 1), *Wo = F32(pi + 2),
                *bo = F32(pi + 3), *g1 = F32(pi + 4), *be1 = F32(pi + 5),
                *W1 = F32(pi + 6), *bl1 = F32(pi + 7), *W2 = F32(pi + 8),
                *bl2 = F32(pi + 9), *g2 = F32(pi + 10), *be2 = F32(pi + 11);
    launch_gemm(0, xb, Wqkv, bqkv, nullptr, 0, t0, M, 1536, 512, stream);
    dim3 ga((unsigned)((Lx + 15) / 16), 8, (unsigned)B);
    attn_kernel<<<ga, 32, 0, stream>>>(t0, t1, Lx);
    launch_gemm(0, t1, Wo, bo, nullptr, 0, t2, M, 512, 512, stream);
    ln_kernel<<<(unsigned)M, 256, 0, stream>>>(t2, xb, g1, be1, xb);
    launch_gemm(1, xb, W1, bl1, nullptr, 0, t0, M, F, 512, stream);
    launch_gemm(0, t0, W2, bl2, nullptr, 0, t2, M, 512, F, stream);
    ln_kernel<<<(unsigned)M, 256, 0, stream>>>(t2, xb, g2, be2, xb);
  };

  for (int layer = 0; layer < NL; ++layer) {
    enc(jamo, BL, L, JL[layer]);
    enc(syl, BS, (int)S, SL[layer]);

    const int ci = CR[layer];
    const float *s2jW1 = F32(ci + 0), *s2jb1 = F32(ci + 1), *s2jW2 = F32(ci + 2),
                *s2jb2 = F32(ci + 3), *j2sW1 = F32(ci + 4), *j2sb1 = F32(ci + 5),
                *j2sW2 = F32(ci + 6), *j2sb2 = F32(ci + 7), *cg1 = F32(ci + 8),
                *cbe1 = F32(ci + 9), *cg2 = F32(ci + 10), *cbe2 = F32(ci + 11);

    // syl_ctx = mlp2(syl); enh_j = jamo + gather(syl_ctx)
    launch_gemm(2, syl, s2jW1, s2jb1, nullptr, 0, t1, BS, 512, 512, stream);
    launch_gemm(0, t1, s2jW2, s2jb2, nullptr, 0, t2, BS, 512, 512, stream);
    gather_kernel<<<gridE(BL * 512), 256, 0, stream>>>(jamo, t2, seg, t0, L, (int)S, BL * 512);
    // jamo_ctx = mlp2(jamo); enh_s = syl + segmean(jamo_ctx)
    launch_gemm(2, jamo, j2sW1, j2sb1, nullptr, 0, t1, BL, 512, 512, stream);
    launch_gemm(0, t1, j2sW2, j2sb2, nullptr, 0, t2, BL, 512, 512, stream);
    hipMemsetAsync(si, 0, BS * 512 * 4, stream);
    segsum_kernel<<<gseg, 256, 0, stream>>>(t2, seg, si, L, (int)S);
    enhs_kernel<<<gridE(BS * 512), 256, 0, stream>>>(syl, si, cnt, t1, BS * 512);
    ln_kernel<<<(unsigned)BL, 256, 0, stream>>>(t0, nullptr, cg1, cbe1, jamo);
    ln_kernel<<<(unsigned)BS, 256, 0, stream>>>(t1, nullptr, cg2, cbe2, syl);
  }

  // ---- outputs ----
  hipMemcpyAsync(outJamo, jamo, BL * 512 * 4, hipMemcpyDeviceToDevice, stream);
  hipMemcpyAsync(outSyl, syl, BS * 512 * 4, hipMemcpyDeviceToDevice, stream);
  launch_gemm(2, jamo, posW1, posb1, nullptr, 0, t0, BL, 256, 512, stream);
  gemm3_kernel<false><<<gridE(BL), 256, 0, stream>>>(t0, posW2, posb2, outJp, (int)BL, 256);
}